// CustomGNNvA_5446018532003
// MI455X (gfx1250) — compile-verified
//
#include <hip/hip_runtime.h>
#include <math.h>

#define HCC 64   // H*C = hidden width (also HS2)
#define HH 4     // heads
#define CCH 16   // channels per head
#define EPSF 1e-5f

typedef float v2f __attribute__((ext_vector_type(2)));
typedef float v8f __attribute__((ext_vector_type(8)));

// ---------------------------------------------------------------------------
// Dense GEMM: Y[nrows x 64] = X[nrows x K] @ W[K x 64] + bvec
// One wave32 computes one 16x16 output tile with V_WMMA_F32_16X16X4_F32.
// Block = 128 threads = 4 waves = the 4 column tiles of the 64-wide output.
// A layout (16x4 f32): lanes 0-15 hold {K=0,K=1}, lanes 16-31 hold {K=2,K=3}
// of rows M=lane&15. B layout mirrors with N=lane&15. C/D: VGPR r holds
// M=r (lanes 0-15) / M=r+8 (lanes 16-31), N=lane&15.
// ---------------------------------------------------------------------------
__global__ __launch_bounds__(128)
void gemm16_wmma_f32(const float* __restrict__ X, const float* __restrict__ W,
                     const float* __restrict__ bvec, float* __restrict__ Y,
                     int nrows, int K) {
  const int wave = threadIdx.x >> 5;
  const int lane = threadIdx.x & 31;
  const int half = lane >> 4;     // 0: K+{0,1}, 1: K+{2,3}
  const int lr   = lane & 15;
  const int row0 = blockIdx.x << 4;
  const int col0 = wave << 4;

  int r = row0 + lr;
  if (r >= nrows) r = nrows - 1;             // clamp (keeps EXEC all-ones)
  const float* __restrict__ xrow = X + (size_t)r * K;

  v8f acc = {};
  for (int k = 0; k < K; k += 4) {
    const int kk = k + (half << 1);
    v2f a, b;
    a.x = xrow[kk];
    a.y = xrow[kk + 1];
    b.x = W[(size_t)kk * HCC + col0 + lr];
    b.y = W[(size_t)(kk + 1) * HCC + col0 + lr];
    __builtin_prefetch(xrow + k + 16, 0, 1);   // global_prefetch_b8
    acc = __builtin_amdgcn_wmma_f32_16x16x4_f32(
        /*neg_a=*/false, a, /*neg_b=*/false, b,
        /*c_mod=*/(short)0, acc, /*reuse_a=*/false, /*reuse_b=*/false);
  }

  const int outc = col0 + lr;
  const float bb = bvec[outc];
#pragma unroll
  for (int q = 0; q < 8; ++q) {
    const int m = row0 + q + (half << 3);
    if (m < nrows) Y[(size_t)m * HCC + outc] = acc[q] + bb;
  }
}

// ---------------------------------------------------------------------------
// Ordered-float encoding so unsigned atomicMax == float max.
// ---------------------------------------------------------------------------
__device__ __forceinline__ unsigned encf(float f) {
  unsigned u = __float_as_uint(f);
  return (u & 0x80000000u) ? ~u : (u | 0x80000000u);
}
__device__ __forceinline__ float decf(unsigned u) {
  return __uint_as_float((u & 0x80000000u) ? (u & 0x7fffffffu) : ~u);
}

// init per-layer accumulators: agg[n,64] = bias[ch]; amax = 0 (minimum key); asum = 0
__global__ void init_layer_k(unsigned* __restrict__ amax, float* __restrict__ asum,
                             float* __restrict__ agg, const float* __restrict__ bias,
                             int n) {
  const int i = blockIdx.x * blockDim.x + threadIdx.x;
  if (i < n * HCC) agg[i] = bias[i & (HCC - 1)];
  if (i < n * HH) { amax[i] = 0u; asum[i] = 0.f; }
}

// attention logits per edge/head + per-destination running max
__global__ void edge_logits_k(const float* __restrict__ xl, const float* __restrict__ xr,
                              const int* __restrict__ src, const int* __restrict__ dst,
                              const float* __restrict__ eattr, const float* __restrict__ We,
                              const float* __restrict__ att,
                              float* __restrict__ elog, unsigned* __restrict__ amax, int E) {
  const int e = blockIdx.x * blockDim.x + threadIdx.x;
  if (e >= E) return;
  const int s = src[e], d = dst[e];
  float ea0 = eattr[(size_t)e * 4 + 0], ea1 = eattr[(size_t)e * 4 + 1];
  float ea2 = eattr[(size_t)e * 4 + 2], ea3 = eattr[(size_t)e * 4 + 3];
  const float* __restrict__ xls = xl + (size_t)s * HCC;
  const float* __restrict__ xrd = xr + (size_t)d * HCC;
#pragma unroll
  for (int h = 0; h < HH; ++h) {
    float acc = 0.f;
#pragma unroll
    for (int c = 0; c < CCH; ++c) {
      const int ch = h * CCH + c;
      float m = xls[ch] + xrd[ch]
              + ea0 * We[ch] + ea1 * We[HCC + ch]
              + ea2 * We[2 * HCC + ch] + ea3 * We[3 * HCC + ch];
      m = m > 0.f ? m : 0.2f * m;                // LeakyReLU(0.2)
      acc = fmaf(m, att[ch], acc);
    }
    elog[(size_t)e * HH + h] = acc;
    atomicMax(&amax[(size_t)d * HH + h], encf(acc));
  }
}

// exp(logit - max) and per-destination sums
__global__ void edge_exp_k(float* __restrict__ elog, const int* __restrict__ dst,
                           const unsigned* __restrict__ amax, float* __restrict__ asum, int E) {
  const int e = blockIdx.x * blockDim.x + threadIdx.x;
  if (e >= E) return;
  const int d = dst[e];
#pragma unroll
  for (int h = 0; h < HH; ++h) {
    const float v = __expf(elog[(size_t)e * HH + h] - decf(amax[(size_t)d * HH + h]));
    elog[(size_t)e * HH + h] = v;
    atomicAdd(&asum[(size_t)d * HH + h], v);
  }
}

// out[dst] += xl[src] * alpha
__global__ void edge_aggregate_k(const float* __restrict__ xl, const float* __restrict__ elog,
                                 const float* __restrict__ asum, const int* __restrict__ src,
                                 const int* __restrict__ dst, float* __restrict__ agg, int E) {
  const int e = blockIdx.x * blockDim.x + threadIdx.x;
  if (e >= E) return;
  const int s = src[e], d = dst[e];
#pragma unroll
  for (int h = 0; h < HH; ++h) {
    const float alpha = elog[(size_t)e * HH + h] / (asum[(size_t)d * HH + h] + 1e-16f);
#pragma unroll
    for (int c = 0; c < CCH; ++c) {
      const int ch = h * CCH + c;
      atomicAdd(&agg[(size_t)d * HCC + ch], xl[(size_t)s * HCC + ch] * alpha);
    }
  }
}

// batch-norm stats: one block per channel (width fixed at 64)
__global__ __launch_bounds__(256)
void bn_stats_k(const float* __restrict__ x, float* __restrict__ mu,
                float* __restrict__ rstd, int n) {
  const int ch = blockIdx.x;
  float s = 0.f, ss = 0.f;
  for (int i = threadIdx.x; i < n; i += 256) {
    const float v = x[(size_t)i * HCC + ch];
    s += v; ss += v * v;
  }
  __shared__ float shs[256], shq[256];
  shs[threadIdx.x] = s; shq[threadIdx.x] = ss;
  __syncthreads();
  for (int off = 128; off > 0; off >>= 1) {
    if (threadIdx.x < off) {
      shs[threadIdx.x] += shs[threadIdx.x + off];
      shq[threadIdx.x] += shq[threadIdx.x + off];
    }
    __syncthreads();
  }
  if (threadIdx.x == 0) {
    const float m = shs[0] / (float)n;
    const float var = shq[0] / (float)n - m * m;  // biased variance
    mu[ch] = m;
    rstd[ch] = rsqrtf(var + EPSF);
  }
}

// apply BN affine + ELU, in place
__global__ void bn_elu_k(float* __restrict__ x, const float* __restrict__ mu,
                         const float* __restrict__ rstd, const float* __restrict__ g,
                         const float* __restrict__ b, int total) {
  const int i = blockIdx.x * blockDim.x + threadIdx.x;
  if (i >= total) return;
  const int ch = i & (HCC - 1);
  const float v = (x[i] - mu[ch]) * rstd[ch] * g[ch] + b[ch];
  x[i] = v > 0.f ? v : expm1f(v);
}

// final head: out[n] = 5 * tanh(x[n,:] @ W2 + b2)
__global__ void head_k(const float* __restrict__ x, const float* __restrict__ W2,
                       const float* __restrict__ b2, float* __restrict__ out, int n) {
  const int i = blockIdx.x * blockDim.x + threadIdx.x;
  if (i >= n) return;
  float acc = b2[0];
#pragma unroll
  for (int c = 0; c < HCC; ++c) acc = fmaf(x[(size_t)i * HCC + c], W2[c], acc);
  out[i] = 5.f * tanhf(acc);
}

// ---------------------------------------------------------------------------
extern "C" void kernel_launch(void* const* d_in, const int* in_sizes, int n_in,
                              void* d_out, int out_size, void* d_ws, size_t ws_size,
                              hipStream_t stream) {
  (void)n_in; (void)out_size; (void)ws_size;

  const float* x0    = (const float*)d_in[0];
  const int*   eidx  = (const int*)  d_in[1];
  const float* eattr = (const float*)d_in[2];
  const float *Wl0 = (const float*)d_in[3],  *bl0 = (const float*)d_in[4];
  const float *Wr0 = (const float*)d_in[5],  *br0 = (const float*)d_in[6];
  const float *We0 = (const float*)d_in[7],  *att0= (const float*)d_in[8];
  const float *bias0=(const float*)d_in[9],  *g0  = (const float*)d_in[10];
  const float *be0 = (const float*)d_in[11];
  const float *Wl1 = (const float*)d_in[12], *bl1 = (const float*)d_in[13];
  const float *Wr1 = (const float*)d_in[14], *br1 = (const float*)d_in[15];
  const float *We1 = (const float*)d_in[16], *att1= (const float*)d_in[17];
  const float *bias1=(const float*)d_in[18], *g1  = (const float*)d_in[19];
  const float *be1 = (const float*)d_in[20];
  const float *W1  = (const float*)d_in[21], *b1  = (const float*)d_in[22];
  const float *gf  = (const float*)d_in[23], *bf  = (const float*)d_in[24];
  const float *W2  = (const float*)d_in[25], *b2  = (const float*)d_in[26];

  const int N = in_sizes[0] / 128;
  const int E = in_sizes[1] / 2;
  const int* src = eidx;
  const int* dst = eidx + E;

  // workspace carve-up
  float* xl   = (float*)d_ws;                       // N x 64
  float* xr   = xl + (size_t)N * HCC;               // N x 64
  float* hid  = xr + (size_t)N * HCC;               // N x 64
  float* elog = hid + (size_t)N * HCC;              // E x 4
  unsigned* amax = (unsigned*)(elog + (size_t)E * HH);  // N x 4
  float* asum = (float*)amax + (size_t)N * HH;      // N x 4
  float* mu   = asum + (size_t)N * HH;              // 64
  float* rstd = mu + HCC;                           // 64

  const dim3 gGemm((N + 15) / 16), bGemm(128);
  const int gEdge   = (E + 255) / 256;
  const int gNode64 = (N * HCC + 255) / 256;
  const int gNode   = (N + 255) / 256;

  // ---------------- Layer 0 (D_IN = 128) ----------------
  gemm16_wmma_f32<<<gGemm, bGemm, 0, stream>>>(x0, Wl0, bl0, xl, N, 128);
  gemm16_wmma_f32<<<gGemm, bGemm, 0, stream>>>(x0, Wr0, br0, xr, N, 128);
  init_layer_k<<<gNode64, 256, 0, stream>>>(amax, asum, hid, bias0, N);
  edge_logits_k<<<gEdge, 256, 0, stream>>>(xl, xr, src, dst, eattr, We0, att0, elog, amax, E);
  edge_exp_k<<<gEdge, 256, 0, stream>>>(elog, dst, amax, asum, E);
  edge_aggregate_k<<<gEdge, 256, 0, stream>>>(xl, elog, asum, src, dst, hid, E);
  bn_stats_k<<<HCC, 256, 0, stream>>>(hid, mu, rstd, N);
  bn_elu_k<<<gNode64, 256, 0, stream>>>(hid, mu, rstd, g0, be0, N * HCC);

  // ---------------- Layer 1 (din = 64) ----------------
  gemm16_wmma_f32<<<gGemm, bGemm, 0, stream>>>(hid, Wl1, bl1, xl, N, 64);
  gemm16_wmma_f32<<<gGemm, bGemm, 0, stream>>>(hid, Wr1, br1, xr, N, 64);
  init_layer_k<<<gNode64, 256, 0, stream>>>(amax, asum, hid, bias1, N);  // hid consumed by gemms
  edge_logits_k<<<gEdge, 256, 0, stream>>>(xl, xr, src, dst, eattr, We1, att1, elog, amax, E);
  edge_exp_k<<<gEdge, 256, 0, stream>>>(elog, dst, amax, asum, E);
  edge_aggregate_k<<<gEdge, 256, 0, stream>>>(xl, elog, asum, src, dst, hid, E);
  bn_stats_k<<<HCC, 256, 0, stream>>>(hid, mu, rstd, N);
  bn_elu_k<<<gNode64, 256, 0, stream>>>(hid, mu, rstd, g1, be1, N * HCC);

  // ---------------- Head ----------------
  gemm16_wmma_f32<<<gGemm, bGemm, 0, stream>>>(hid, W1, b1, xl, N, 64);  // reuse xl as h1
  bn_stats_k<<<HCC, 256, 0, stream>>>(xl, mu, rstd, N);
  bn_elu_k<<<gNode64, 256, 0, stream>>>(xl, mu, rstd, gf, bf, N * HCC);
  head_k<<<gNode, 256, 0, stream>>>(xl, W2, b2, (float*)d_out, N);
}